// RGCN_21096879358045
// MI455X (gfx1250) — compile-verified
//
#include <hip/hip_runtime.h>

// ---------------------------------------------------------------------------
// RGCN (2 layers, basis decomposition, per-relation mean aggregation)
// MI455X strategy:
//   * W1 = comp1 @ bases1            (tiny kernel)
//   * XW1[r] = x @ W1[r]             (f32 WMMA 16x16x4, 8 relations)
//   * h1     = x @ root1 + bias1     (f32 WMMA 16x16x4)
//   * per-(r,dst) edge counts -> reciprocal (shared by both layers)
//   * edge pass 1: h1[dst] += (1/cnt[r,dst]) * XW1[r][src]   (f32 atomics,
//       38MB accumulator resident in 192MB L2)
//   * layer 2 collapses to column 1: per-node 9 dot64 products, then 4M
//       scalar atomics into d_out.
// N = 150000 is an exact multiple of 16, so the WMMA tiles need no edge
// predication: stores are straight-line (no exec-mask toggling).
// ---------------------------------------------------------------------------

typedef __attribute__((ext_vector_type(2))) float v2f;
typedef __attribute__((ext_vector_type(8))) float v8f;

#define ND 100000
#define NP 50000
#define NN 150000
#define RR 8
#define BB 30
#define HH 64

// ---------------- tiny weight-build kernels --------------------------------

// W[r, p] = sum_b comp[r,b] * bases[b, p]   (p in [0, io))
__global__ void build_w_kernel(const float* __restrict__ comp,
                               const float* __restrict__ bases,
                               float* __restrict__ W, int nmat, int nb, int io)
{
    int t = blockIdx.x * blockDim.x + threadIdx.x;
    if (t >= nmat * io) return;
    int r = t / io, p = t % io;
    float acc = 0.f;
    for (int b = 0; b < nb; ++b)
        acc += comp[r * nb + b] * bases[(size_t)b * io + p];
    W[t] = acc;
}

// w2c[r*64+k] = sum_b comp2[r,b] * bases2[b, k, 1]   (column 1 only)
__global__ void build_w2col_kernel(const float* __restrict__ comp,
                                   const float* __restrict__ bases,
                                   float* __restrict__ w)
{
    int t = blockIdx.x * blockDim.x + threadIdx.x;
    if (t >= RR * HH) return;
    int r = t / HH, k = t % HH;
    float acc = 0.f;
    for (int b = 0; b < BB; ++b)
        acc += comp[r * BB + b] * bases[b * HH * 2 + k * 2 + 1];
    w[t] = acc;
}

// ---------------- counts ---------------------------------------------------

__global__ void zero_kernel(float* __restrict__ p, int n)
{
    int t = blockIdx.x * blockDim.x + threadIdx.x;
    if (t < n) p[t] = 0.f;
}

__global__ void count_kernel(const int* __restrict__ dst,
                             const int* __restrict__ etype,
                             float* __restrict__ cnt, int E)
{
    int e = blockIdx.x * blockDim.x + threadIdx.x;
    if (e >= E) return;
    atomicAdd(&cnt[(size_t)etype[e] * NN + dst[e]], 1.0f);
}

__global__ void recip_kernel(float* __restrict__ cnt, int n)
{
    int t = blockIdx.x * blockDim.x + threadIdx.x;
    if (t < n) cnt[t] = 1.0f / fmaxf(cnt[t], 1.0f);
}

// ---------------- dense node GEMM via f32 WMMA -----------------------------
// out[mat][n][0..63] = x[n] @ W[mat]  (+ bias), x gathered from embeddings.
// One wave handles one (16-node tile, mat). 16 K-steps of V_WMMA_F32_16X16X4.
// A layout (16x4 f32): lanes 0-15 hold M=0..15; VGPR0 = K={0|2}, VGPR1 = K+1.
// B layout (4x16):     VGPR0 row K={0|2} striped over lanes, VGPR1 = K+1.
// C/D layout: VGPR v -> M = v + 8*(lane>=16), N = lane&15.
__global__ void node_gemm_wmma(const int* __restrict__ xd, const int* __restrict__ xp,
                               const float* __restrict__ embd,
                               const float* __restrict__ embp,
                               const float* __restrict__ W,    // [numMat][64][64]
                               float* __restrict__ out,        // [numMat][NN][64]
                               const float* __restrict__ bias, // nullable [64]
                               int numMat, int numTiles)
{
    const int lane = threadIdx.x & 31;
    const int wave = threadIdx.x >> 5;
    const int tilesPerBlock = 8 / numMat;
    const int mat  = wave % numMat;
    const int tile = blockIdx.x * tilesPerBlock + wave / numMat;
    if (tile >= numTiles) return;   // wave-uniform; interior tiles are exact

    const int half = lane >> 4;   // which K pair / which M half for stores
    const int m    = lane & 15;

    const int node = tile * 16 + m;   // always < NN (NN % 16 == 0)
    const float* row = (node < ND) ? (embd + (size_t)xd[node] * HH)
                                   : (embp + (size_t)xp[node - ND] * HH);
    const v2f* row2 = (const v2f*)row;

    // Preload all 16 A fragments (32 VGPRs), reuse across 4 column tiles.
    v2f a[16];
#pragma unroll
    for (int kk = 0; kk < 16; ++kk) a[kk] = row2[2 * kk + half];

    const float* Wm   = W   + (size_t)mat * HH * HH;
    float*       outm = out + (size_t)mat * (size_t)NN * HH;
    float*       orow = outm + (size_t)(tile * 16 + half * 8) * HH;

#pragma unroll
    for (int c = 0; c < 4; ++c) {
        const int col = c * 16 + m;
        const float binit = bias ? bias[col] : 0.f;
        v8f acc;
#pragma unroll
        for (int v = 0; v < 8; ++v) acc[v] = binit;

#pragma unroll
        for (int kk = 0; kk < 16; ++kk) {
            v2f b;
            b.x = Wm[(4 * kk + 2 * half + 0) * HH + col];
            b.y = Wm[(4 * kk + 2 * half + 1) * HH + col];
            acc = __builtin_amdgcn_wmma_f32_16x16x4_f32(
                /*neg_a=*/false, a[kk], /*neg_b=*/false, b,
                /*c_mod=*/(short)0, acc, /*reuse_a=*/false, /*reuse_b=*/false);
        }

        // Straight-line stores: lane L and row v -> (node tile*16+half*8+v, col)
#pragma unroll
        for (int v = 0; v < 8; ++v)
            orow[(size_t)v * HH + col] = acc[v];
    }
}

// ---------------- edge aggregation, layer 1 --------------------------------
// 16 threads per edge; each handles 4 contiguous channels (float4 gather,
// 4 float atomics into L2-resident h1).
__global__ void edge1_kernel(const int* __restrict__ src, const int* __restrict__ dst,
                             const int* __restrict__ etype,
                             const float* __restrict__ cinv,
                             const float* __restrict__ XW1,
                             float* __restrict__ h1, int E)
{
    long t = (long)blockIdx.x * blockDim.x + threadIdx.x;
    long e = t >> 4;
    if (e >= E) return;
    int kq = (int)(t & 15) * 4;
    int s = src[e], d = dst[e], r = etype[e];
    float scale = cinv[(size_t)r * NN + d];
    const float4 v = *(const float4*)(XW1 + ((size_t)r * NN + s) * HH + kq);
    float* hp = h1 + (size_t)d * HH + kq;
    atomicAdd(hp + 0, scale * v.x);
    atomicAdd(hp + 1, scale * v.y);
    atomicAdd(hp + 2, scale * v.z);
    atomicAdd(hp + 3, scale * v.w);
}

// ---------------- layer-2 per-node (messages column 1 + root init) ---------
__global__ void l2_node_kernel(const float* __restrict__ h1,
                               const float* __restrict__ w2c,   // [R][64]
                               const float* __restrict__ root2, // [64][2]
                               const float* __restrict__ bias2, // [2]
                               float* __restrict__ xw2,         // [R][NN]
                               float* __restrict__ out)         // [NN]
{
    int n = blockIdx.x * blockDim.x + threadIdx.x;
    if (n >= NN) return;
    const float* hrow = h1 + (size_t)n * HH;
    float acc[RR + 1];
#pragma unroll
    for (int r = 0; r <= RR; ++r) acc[r] = 0.f;
    for (int k = 0; k < HH; ++k) {
        float v = hrow[k];
#pragma unroll
        for (int r = 0; r < RR; ++r) acc[r] += v * w2c[r * HH + k];
        acc[RR] += v * root2[k * 2 + 1];
    }
#pragma unroll
    for (int r = 0; r < RR; ++r) xw2[(size_t)r * NN + n] = acc[r];
    out[n] = acc[RR] + bias2[1];
}

// ---------------- edge aggregation, layer 2 (scalar) -----------------------
__global__ void edge2_kernel(const int* __restrict__ src, const int* __restrict__ dst,
                             const int* __restrict__ etype,
                             const float* __restrict__ cinv,
                             const float* __restrict__ xw2,
                             float* __restrict__ out, int E)
{
    int e = blockIdx.x * blockDim.x + threadIdx.x;
    if (e >= E) return;
    int r = etype[e], d = dst[e];
    atomicAdd(&out[d], cinv[(size_t)r * NN + d] * xw2[(size_t)r * NN + src[e]]);
}

// ---------------------------------------------------------------------------

extern "C" void kernel_launch(void* const* d_in, const int* in_sizes, int n_in,
                              void* d_out, int out_size, void* d_ws, size_t ws_size,
                              hipStream_t stream)
{
    const int*   xd     = (const int*)d_in[0];
    const int*   xp     = (const int*)d_in[1];
    const int*   ei     = (const int*)d_in[2];
    const int*   et     = (const int*)d_in[3];
    const float* embd   = (const float*)d_in[4];
    const float* embp   = (const float*)d_in[5];
    const float* comp1  = (const float*)d_in[6];
    const float* bases1 = (const float*)d_in[7];
    const float* root1  = (const float*)d_in[8];
    const float* bias1  = (const float*)d_in[9];
    const float* comp2  = (const float*)d_in[10];
    const float* bases2 = (const float*)d_in[11];
    const float* root2  = (const float*)d_in[12];
    const float* bias2  = (const float*)d_in[13];

    const int E = in_sizes[3];
    const int* src = ei;
    const int* dst = ei + E;

    // Workspace carve (floats, 256B aligned slices)
    float* ws = (float*)d_ws;
    size_t off = 0;
    auto carve = [&](size_t n) {
        float* p = ws + off;
        off += (n + 63) & ~(size_t)63;
        return p;
    };
    float* W1   = carve((size_t)RR * HH * HH);     // 128 KB
    float* w2c  = carve((size_t)RR * HH);          // 2 KB
    float* cinv = carve((size_t)RR * NN);          // 4.8 MB
    float* XW1  = carve((size_t)RR * NN * HH);     // 307 MB
    float* h1   = carve((size_t)NN * HH);          // 38.4 MB
    float* xw2  = carve((size_t)RR * NN);          // 4.8 MB
    float* out  = (float*)d_out;

    const int TPB = 256;
    const int numTiles = NN / 16;                  // 9375 (exact)

    // 1) Materialize relation weights
    build_w_kernel<<<(RR * HH * HH + TPB - 1) / TPB, TPB, 0, stream>>>(
        comp1, bases1, W1, RR, BB, HH * HH);
    build_w2col_kernel<<<(RR * HH + TPB - 1) / TPB, TPB, 0, stream>>>(
        comp2, bases2, w2c);

    // 2) Per-(relation, dst) edge counts -> reciprocals (shared by both layers)
    zero_kernel<<<(RR * NN + TPB - 1) / TPB, TPB, 0, stream>>>(cinv, RR * NN);
    count_kernel<<<(E + TPB - 1) / TPB, TPB, 0, stream>>>(dst, et, cinv, E);
    recip_kernel<<<(RR * NN + TPB - 1) / TPB, TPB, 0, stream>>>(cinv, RR * NN);

    // 3) Dense WMMA GEMMs: XW1[r] = x @ W1[r];  h1 = x @ root1 + bias1
    node_gemm_wmma<<<numTiles, TPB, 0, stream>>>(
        xd, xp, embd, embp, W1, XW1, nullptr, RR, numTiles);
    node_gemm_wmma<<<(numTiles + 7) / 8, TPB, 0, stream>>>(
        xd, xp, embd, embp, root1, h1, bias1, 1, numTiles);

    // 4) Layer-1 edge aggregation: h1[dst] += XW1[r][src] / cnt[r,dst]
    long t1 = (long)E * 16;
    edge1_kernel<<<(unsigned)((t1 + TPB - 1) / TPB), TPB, 0, stream>>>(
        src, dst, et, cinv, XW1, h1, E);

    // 5) Layer 2 (column 1 only): per-node dots, then scalar edge atomics
    l2_node_kernel<<<(NN + TPB - 1) / TPB, TPB, 0, stream>>>(
        h1, w2c, root2, bias2, xw2, out);
    edge2_kernel<<<(E + TPB - 1) / TPB, TPB, 0, stream>>>(
        src, dst, et, cinv, xw2, out, E);
}